// BackboneGNN2_63316407878053
// MI455X (gfx1250) — compile-verified
//
#include <hip/hip_runtime.h>
#include <hip/hip_bf16.h>

typedef __attribute__((ext_vector_type(2))) float v2f;
typedef __attribute__((ext_vector_type(8))) float v8f;

#define N_NODES   50000
#define N_EDGES   600000
#define IN_DIM    128
#define HID       128
#define N_CLASSES 40
#define N_GRAPHS  128

// ---------------------------------------------------------------- fill zero
__global__ void fill_zero_kernel(float* __restrict__ p, int n) {
    int i = blockIdx.x * blockDim.x + threadIdx.x;
    if (i < n) p[i] = 0.0f;
}

// ------------------------------------------------------- edge gather/scatter
// One wave32 per edge: lanes cover 128 dims as 4 floats each (b128 load).
// message = relu(h[src]); mean numerator into agg[dst] via f32 atomics,
// denominator into cnt[dst].
__global__ __launch_bounds__(256)
void sage_scatter_kernel(const float* __restrict__ h,
                         const int*  __restrict__ src,
                         const int*  __restrict__ dst,
                         float* __restrict__ agg,
                         float* __restrict__ cnt,
                         int n_edges) {
    int wid  = (blockIdx.x * blockDim.x + threadIdx.x) >> 5;
    int lane = threadIdx.x & 31;
    if (wid >= n_edges) return;
    int s = src[wid];
    int d = dst[wid];
    const float4 v = *(const float4*)(h + (size_t)s * IN_DIM + lane * 4);
    float* ad = agg + (size_t)d * IN_DIM + lane * 4;
    atomicAdd(ad + 0, fmaxf(v.x, 0.0f));
    atomicAdd(ad + 1, fmaxf(v.y, 0.0f));
    atomicAdd(ad + 2, fmaxf(v.z, 0.0f));
    atomicAdd(ad + 3, fmaxf(v.w, 0.0f));
    if (lane == 0) atomicAdd(cnt + d, 1.0f);
}

// ----------------------------------------------------------------- WMMA GEMM
// hout[m,n] = relu?( sum_k (agg[m,k]/max(cnt,1))*Wl[n,k]
//                  + sum_k hin[m,k]*Wr[n,k] + bl[n] )
// One wave per 16x16 tile, V_WMMA_F32_16X16X4_F32 over K=128 twice.
// A layout (16x4 f32): lane<16 -> K={k,k+1}, lane>=16 -> K={k+2,k+3}, M=lane&15
// B layout (4x16 f32): same K-pair split per half-wave, N=lane&15
// Out-of-range output columns (last layer, dout=40) read a zeroed dummy row
// instead of masking with a multiply -> pure load+WMMA inner loop.
__global__ __launch_bounds__(128)
void sage_gemm_kernel(const float* __restrict__ agg,
                      const float* __restrict__ cnt,
                      const float* __restrict__ hin,
                      const float* __restrict__ Wl,
                      const float* __restrict__ bl,
                      const float* __restrict__ Wr,
                      const float* __restrict__ zrow,
                      float* __restrict__ hout,
                      int n_nodes, int dout, int do_relu) {
    const int lane = threadIdx.x & 31;
    const int wave = threadIdx.x >> 5;
    const int row_tile = blockIdx.x * 4 + wave;
    if (row_tile * 16 >= n_nodes) return;          // wave-uniform exit

    const int m    = row_tile * 16 + (lane & 15);  // node row (A)
    const int half = lane >> 4;                    // K-pair selector
    const int n    = blockIdx.y * 16 + (lane & 15);// out col (B)
    const bool ncol_ok = (n < dout);

    const float inv = 1.0f / fmaxf(cnt[m], 1.0f);

    const float* Arow0 = agg + (size_t)m * IN_DIM;
    const float* Arow1 = hin + (size_t)m * IN_DIM;
    const float* Brow0 = ncol_ok ? (Wl + (size_t)n * IN_DIM) : zrow;
    const float* Brow1 = ncol_ok ? (Wr + (size_t)n * IN_DIM) : zrow;

    v8f acc = {};
    // Phase 0: mean-aggregated neighbors vs Wl (divide folded into A load)
    #pragma unroll 8
    for (int k = 0; k < IN_DIM; k += 4) {
        const int kk = k + 2 * half;
        v2f av = *(const v2f*)(Arow0 + kk);
        v2f bv = *(const v2f*)(Brow0 + kk);
        av *= inv;
        acc = __builtin_amdgcn_wmma_f32_16x16x4_f32(
            false, av, false, bv, (short)0, acc, false, false);
    }
    // Phase 1: self features vs Wr (pure load + WMMA)
    #pragma unroll 8
    for (int k = 0; k < IN_DIM; k += 4) {
        const int kk = k + 2 * half;
        v2f av = *(const v2f*)(Arow1 + kk);
        v2f bv = *(const v2f*)(Brow1 + kk);
        acc = __builtin_amdgcn_wmma_f32_16x16x4_f32(
            false, av, false, bv, (short)0, acc, false, false);
    }

    // C/D layout: VGPR r -> row (r + 8*half), col = lane&15
    const float bias = ncol_ok ? bl[n] : 0.0f;
    const int row_base = row_tile * 16 + half * 8;
    #pragma unroll
    for (int r = 0; r < 8; ++r) {
        float v = acc[r] + bias;
        if (do_relu) v = fmaxf(v, 0.0f);
        const int row = row_base + r;
        if (ncol_ok && row < n_nodes)
            hout[(size_t)row * dout + n] = v;
    }
}

// -------------------------------------------------------------- mean pooling
__global__ __launch_bounds__(256)
void pool_accum_kernel(const float* __restrict__ h,
                       const int*   __restrict__ batch,
                       float* __restrict__ gsum,
                       float* __restrict__ gcnt,
                       int n_nodes, int dout) {
    int node = (blockIdx.x * blockDim.x + threadIdx.x) >> 5;
    int lane = threadIdx.x & 31;
    if (node >= n_nodes) return;
    int b = batch[node];
    const float* hr = h + (size_t)node * dout;
    float* gs = gsum + (size_t)b * dout;
    for (int d = lane; d < dout; d += 32)
        atomicAdd(gs + d, hr[d]);
    if (lane == 0) atomicAdd(gcnt + b, 1.0f);
}

__global__ void pool_final_kernel(const float* __restrict__ gsum,
                                  const float* __restrict__ gcnt,
                                  float* __restrict__ g,
                                  int n_graphs, int dout) {
    int i = blockIdx.x * blockDim.x + threadIdx.x;
    if (i < n_graphs * dout)
        g[i] = gsum[i] / fmaxf(gcnt[i / dout], 1.0f);
}

// ------------------------------------------------------------------- launch
extern "C" void kernel_launch(void* const* d_in, const int* in_sizes, int n_in,
                              void* d_out, int out_size, void* d_ws, size_t ws_size,
                              hipStream_t stream) {
    const float* x     = (const float*)d_in[0];
    const int*   ei    = (const int*)d_in[1];          // [2, E]
    const int*   src   = ei;
    const int*   dst   = ei + N_EDGES;
    const int*   batch = (const int*)d_in[2];
    const float* Wl[3] = {(const float*)d_in[3], (const float*)d_in[6], (const float*)d_in[9]};
    const float* bl[3] = {(const float*)d_in[4], (const float*)d_in[7], (const float*)d_in[10]};
    const float* Wr[3] = {(const float*)d_in[5], (const float*)d_in[8], (const float*)d_in[11]};

    float* out_h = (float*)d_out;                       // [N_NODES, 40]
    float* out_g = out_h + (size_t)N_NODES * N_CLASSES; // [N_GRAPHS, 40]

    // workspace carve-up (floats): agg | cnt | h1 | h2 | gsum | gcnt | zrow
    float* agg  = (float*)d_ws;
    float* cnt  = agg + (size_t)N_NODES * IN_DIM;       // contiguous after agg
    float* h1   = cnt + N_NODES;
    float* h2   = h1 + (size_t)N_NODES * HID;
    float* gsum = h2 + (size_t)N_NODES * HID;
    float* gcnt = gsum + N_GRAPHS * N_CLASSES;          // contiguous after gsum
    float* zrow = gcnt + N_GRAPHS;                      // 128-float zero row

    struct Layer { const float* in; float* out; int dout; int relu; };
    Layer L[3] = { {x,  h1,   HID,       1},
                   {h1, h2,   HID,       1},
                   {h2, out_h, N_CLASSES, 0} };

    const int zero_n = N_NODES * IN_DIM + N_NODES;      // agg + cnt together
    const int row_tiles = (N_NODES + 15) / 16;          // 3125

    // zero the dummy B row once per launch (deterministic; ws is not trusted)
    fill_zero_kernel<<<1, 128, 0, stream>>>(zrow, IN_DIM);

    for (int l = 0; l < 3; ++l) {
        fill_zero_kernel<<<(zero_n + 255) / 256, 256, 0, stream>>>(agg, zero_n);
        sage_scatter_kernel<<<(N_EDGES + 7) / 8, 256, 0, stream>>>(
            L[l].in, src, dst, agg, cnt, N_EDGES);
        dim3 grid((row_tiles + 3) / 4, (L[l].dout + 15) / 16);
        sage_gemm_kernel<<<grid, 128, 0, stream>>>(
            agg, cnt, L[l].in, Wl[l], bl[l], Wr[l], zrow,
            L[l].out, N_NODES, L[l].dout, L[l].relu);
    }

    const int gz = N_GRAPHS * N_CLASSES + N_GRAPHS;     // gsum + gcnt together
    fill_zero_kernel<<<(gz + 255) / 256, 256, 0, stream>>>(gsum, gz);
    pool_accum_kernel<<<(N_NODES * 32 + 255) / 256, 256, 0, stream>>>(
        out_h, batch, gsum, gcnt, N_NODES, N_CLASSES);
    pool_final_kernel<<<(N_GRAPHS * N_CLASSES + 255) / 256, 256, 0, stream>>>(
        gsum, gcnt, out_g, N_GRAPHS, N_CLASSES);
}